// ScaleSpaceAffinePatchExtractor_76536317215047
// MI455X (gfx1250) — compile-verified
//
#include <hip/hip_runtime.h>
#include <math.h>

// ---------------- constants ----------------
#define BORDER   16
#define K_FEAT   2000
#define MR_SIZE  3.0f
#define TX       128                 // 8 waves * 16 cols per block
#define CAND_MAX (1 << 20)
#define TIE_MAX  4096

typedef __attribute__((ext_vector_type(2))) float f32x2;
typedef __attribute__((ext_vector_type(8))) float f32x8;
typedef __attribute__((ext_vector_type(4))) int   hd_v4i;

#if defined(__AMDGCN__) && __has_builtin(__builtin_amdgcn_global_load_async_to_lds_b128)
#define HD_ASYNC 1
#else
#define HD_ASYNC 0
#endif

#if HD_ASYNC
// async copy of 16B global -> LDS; tracked by ASYNCcnt
__device__ __forceinline__ void hd_async_b128(const float* g, float* l) {
  __builtin_amdgcn_global_load_async_to_lds_b128(
      (__attribute__((address_space(1))) hd_v4i*)g,
      (__attribute__((address_space(3))) hd_v4i*)l,
      0, 0);
}
#endif

// sig[i] = 1.6*2^(i/3); delta[i]=sqrt(sig[i]^2-1.6^2)
// coeff table rows: row0 = init sigma blur, rows1..4 = per-level deltas (zero padded to 32)
__global__ void hd_prep_coeffs(float* __restrict__ coef) {
  if (threadIdx.x == 0 && blockIdx.x == 0) {
    const float sig[5] = {1.6f, 1.2262414f, 1.9725057f, 2.7712813f, 3.7006738f};
    const int   rad[5] = {5, 4, 6, 9, 12};
    for (int kk = 0; kk < 5; ++kk) {
      float tmp[32];
      float s = 0.f;
      int r = rad[kk];
      for (int t = 0; t <= 2*r; ++t) {
        float xx = (float)(t - r) / sig[kk];
        tmp[t] = __expf(-0.5f * xx * xx);
        s += tmp[t];
      }
      for (int t = 0; t < 32; ++t)
        coef[kk*32 + t] = (t <= 2*r) ? tmp[t] / s : 0.f;
    }
  }
}

__global__ void hd_init_ws(unsigned* __restrict__ scal, unsigned* __restrict__ hist) {
  int i = blockIdx.x * blockDim.x + threadIdx.x;
  if (i < 16)   scal[i] = 0u;
  if (i < 2048) hist[i] = 0u;
}

// store 16x16 f32 C/D tile (8 VGPRs) with minimal control flow
__device__ __forceinline__ void hd_store_tile(
    float* __restrict__ dst, const f32x8& acc,
    int W, int H, int x, int by, int mb)
{
  if (by + 16 <= H) {            // uniform: all 8 rows in range (H is a multiple of 16)
    if (x < W) {
      float* p = dst + (size_t)(by + mb) * W + x;
      #pragma unroll
      for (int v = 0; v < 8; ++v) { *p = acc[v]; p += W; }
    }
  } else if (x < W) {
    #pragma unroll
    for (int v = 0; v < 8; ++v) {
      int y = by + mb + v;
      if (y < H) dst[(size_t)y * W + x] = acc[v];
    }
  }
}

// ---------------- horizontal blur: out = In(16xL) x Band(Lx16), WMMA f32 16x16x4 ----
// Footprint anchored at bx-16 (64B aligned), width 160, LDS stride 164 (gcd(164,64)=4).
template<int R>
__global__ __launch_bounds__(256) void hd_hblur_wmma(
    const float* __restrict__ src, float* __restrict__ dst,
    int W, int H, const float* __restrict__ coef)
{
  constexpr int FWr  = TX + 32;             // 160
  constexpr int FWSr = FWr + 4;             // 164
  constexpr int NCH  = (16 + 2*R + 3) >> 2;
  __shared__ float sT[16][FWSr];
  __shared__ float sKext[64];               // zero-extended band coefficients
  const int tid  = threadIdx.x;
  const int lane = tid & 31;
  const int wave = tid >> 5;
  const int bx = blockIdx.x * TX;
  const int by = blockIdx.y * 16;
  if (tid < 64) {
    int t = tid - 16;
    sKext[tid] = ((unsigned)t < 32u) ? coef[t] : 0.f;
  }
  const int gx0 = bx - 16;
#if HD_ASYNC
  const bool interior = (gx0 >= 0) && (bx + 144 <= W) && (by + 16 <= H);
  if (interior) {
    for (int i = tid; i < 16 * (FWr/4); i += 256) {
      int row = i / (FWr/4);
      int c4  = (i - row * (FWr/4)) * 4;
      hd_async_b128(src + (size_t)(by + row) * W + gx0 + c4, &sT[row][c4]);
    }
    asm volatile("s_wait_asynccnt 0" ::: "memory");
  } else
#endif
  {
    for (int i = tid; i < 16 * FWr; i += 256) {
      int row = i / FWr, col = i - row * FWr;
      int gx = gx0 + col, gy = by + row;
      float v = 0.f;
      if ((unsigned)gx < (unsigned)W && (unsigned)gy < (unsigned)H)
        v = src[(size_t)gy * W + gx];
      sT[row][col] = v;
    }
  }
  __syncthreads();

  const int n  = lane & 15;        // A row m == B col n per lane
  const int kh = (lane >> 4) << 1; // K sub-offset: 0 / 2
  const int cbase = wave * 16 + 16 - R;
  f32x8 acc = {0.f,0.f,0.f,0.f,0.f,0.f,0.f,0.f};
  #pragma unroll
  for (int c = 0; c < NCH; ++c) {
    const int jb = 4*c + kh;
    f32x2 a, b;
    a.x = sT[n][cbase + jb];
    a.y = sT[n][cbase + jb + 1];
    b.x = sKext[jb - n + 16];
    b.y = sKext[jb - n + 17];
    acc = __builtin_amdgcn_wmma_f32_16x16x4_f32(false, a, false, b, (short)0, acc, false, false);
  }
  hd_store_tile(dst, acc, W, H, bx + wave * 16 + n, by, (lane >> 4) * 8);
}

// ---------------- vertical blur: out = Band(16xL) x In(Lx16) ----------------
template<int R>
__global__ __launch_bounds__(256) void hd_vblur_wmma(
    const float* __restrict__ src, float* __restrict__ dst,
    int W, int H, const float* __restrict__ coef)
{
  constexpr int FHr  = (19 + 2*R + 3) & ~3;
  constexpr int LSTR = 136;                 // 2*LSTR%64==16: lane halves hit disjoint banks
  constexpr int NCH  = (16 + 2*R + 3) >> 2;
  __shared__ float sT[FHr][LSTR];
  __shared__ float sKext[64];
  const int tid  = threadIdx.x;
  const int lane = tid & 31;
  const int wave = tid >> 5;
  const int bx = blockIdx.x * TX;
  const int by = blockIdx.y * 16;
  if (tid < 64) {
    int t = tid - 16;
    sKext[tid] = ((unsigned)t < 32u) ? coef[t] : 0.f;
  }
#if HD_ASYNC
  const bool interior = (by - R >= 0) && (by - R + FHr <= H) && (bx + TX <= W);
  if (interior) {
    for (int i = tid; i < FHr * (TX/4); i += 256) {
      int row = i >> 5;                 // TX/4 == 32 chunks per row
      int c4  = (i & 31) * 4;
      hd_async_b128(src + (size_t)(by - R + row) * W + bx + c4, &sT[row][c4]);
    }
    asm volatile("s_wait_asynccnt 0" ::: "memory");
  } else
#endif
  {
    for (int i = tid; i < FHr * TX; i += 256) {
      int row = i / TX, col = i - row * TX;
      int gx = bx + col, gy = by - R + row;
      float v = 0.f;
      if ((unsigned)gx < (unsigned)W && (unsigned)gy < (unsigned)H)
        v = src[(size_t)gy * W + gx];
      sT[row][col] = v;
    }
  }
  __syncthreads();

  const int n  = lane & 15;
  const int kh = (lane >> 4) << 1;
  const int xw = wave * 16;
  f32x8 acc = {0.f,0.f,0.f,0.f,0.f,0.f,0.f,0.f};
  #pragma unroll
  for (int c = 0; c < NCH; ++c) {
    const int jb = 4*c + kh;
    f32x2 a, b;
    a.x = sKext[jb - n + 16];   // A[m][j] = k[j-m], m == n per lane
    a.y = sKext[jb - n + 17];
    b.x = sT[jb][xw + n];
    b.y = sT[jb + 1][xw + n];
    acc = __builtin_amdgcn_wmma_f32_16x16x4_f32(false, a, false, b, (short)0, acc, false, false);
  }
  hd_store_tile(dst, acc, W, H, bx + xw + n, by, (lane >> 4) * 8);
}

// ---------------- scale-normalized det-Hessian response ----------------
__global__ __launch_bounds__(256) void hd_hess_resp(
    const float* __restrict__ s, float* __restrict__ d, int W, int H, float s4)
{
  const int x = blockIdx.x * 16 + (threadIdx.x & 15);
  const int y = blockIdx.y * 16 + (threadIdx.x >> 4);
  if (x >= W || y >= H) return;
  float c, gxx, gyy, gxy;
  const bool interior = (blockIdx.x > 0) && (blockIdx.y > 0) &&
                        ((int)(blockIdx.x + 1) * 16 < W) && ((int)(blockIdx.y + 1) * 16 < H);
  if (interior) {
    const float* p  = s + (size_t)y * W + x;
    const float* pm = p - W;
    const float* pp = p + W;
    c   = p[0];
    gxx = p[-1] - 2.f * c + p[1];
    gyy = pm[0] - 2.f * c + pp[0];
    gxy = 0.25f * (pm[-1] - pm[1] - pp[-1] + pp[1]);
  } else {
    auto ld = [&](int xx, int yy) -> float {
      if ((unsigned)xx >= (unsigned)W || (unsigned)yy >= (unsigned)H) return 0.f;
      return s[(size_t)yy * W + xx];
    };
    c   = ld(x, y);
    gxx = ld(x-1, y) - 2.f * c + ld(x+1, y);
    gyy = ld(x, y-1) - 2.f * c + ld(x, y+1);
    gxy = 0.25f * (ld(x-1,y-1) - ld(x+1,y-1) - ld(x-1,y+1) + ld(x+1,y+1));
  }
  float v = (gxx * gyy - gxy * gxy) * s4;
  d[(size_t)y * W + x] = fmaxf(v, 0.f);
}

// ---------------- 3x3x3 NMS over levels 1..3 + candidate compaction ----------------
__global__ __launch_bounds__(256) void hd_nms_compact(
    const float* __restrict__ resp, size_t lstride, int W, int H,
    float pix, float s1, float s2, float s3,
    float4* __restrict__ cand, unsigned* __restrict__ cnt)
{
  int x = BORDER + blockIdx.x * 16 + (threadIdx.x & 15);
  int y = BORDER + blockIdx.y * 16 + (threadIdx.x >> 4);
  if (x >= W - BORDER || y >= H - BORDER) return;
  const float ss[3] = {s1, s2, s3};
  for (int li = 1; li <= 3; ++li) {
    const float* rc = resp + (size_t)li * lstride;
    float cur = rc[(size_t)y * W + x];
    if (cur <= 0.f) continue;
    float mx = -1.f;
    #pragma unroll
    for (int dz = -1; dz <= 1; ++dz) {
      const float* rp = resp + (size_t)(li + dz) * lstride;
      #pragma unroll
      for (int dy = -1; dy <= 1; ++dy) {
        const float* row = rp + (size_t)(y + dy) * W + x;
        mx = fmaxf(mx, fmaxf(fmaxf(row[-1], row[0]), row[1]));
      }
    }
    if (cur >= mx) {
      unsigned i = atomicAdd(cnt, 1u);
      if (i < (unsigned)CAND_MAX)
        cand[i] = make_float4(cur, (float)x * pix, (float)y * pix, ss[li - 1]);
    }
  }
}

__global__ __launch_bounds__(256) void hd_downsample(
    const float* __restrict__ s, float* __restrict__ d, int Wd, int Hd, int Ws)
{
  int x = blockIdx.x * 16 + (threadIdx.x & 15);
  int y = blockIdx.y * 16 + (threadIdx.x >> 4);
  if (x < Wd && y < Hd) d[(size_t)y * Wd + x] = s[(size_t)(2 * y) * Ws + 2 * x];
}

// ---------------- top-k: histogram radix select on float bit pattern ----------------
__global__ void hd_hist(const float4* __restrict__ cand, const unsigned* __restrict__ scal,
                        unsigned* __restrict__ hist)
{
  unsigned n = scal[0]; if (n > CAND_MAX) n = CAND_MAX;
  for (unsigned i = blockIdx.x * blockDim.x + threadIdx.x; i < n; i += gridDim.x * blockDim.x) {
    unsigned bin = __float_as_uint(cand[i].x) >> 20;  // values > 0 -> bin in [0,2047]
    atomicAdd(&hist[bin], 1u);
  }
}

__global__ void hd_thresh(const unsigned* __restrict__ hist, unsigned* __restrict__ scal) {
  if (threadIdx.x == 0 && blockIdx.x == 0) {
    unsigned cum = 0; int T = 0; unsigned above = 0;
    for (int b = 2047; b >= 0; --b) {
      unsigned h = hist[b];
      if (cum + h >= (unsigned)K_FEAT || b == 0) { T = b; above = cum; break; }
      cum += h;
    }
    scal[1] = (unsigned)T;
    scal[2] = above;
    scal[3] = ((unsigned)K_FEAT > above) ? ((unsigned)K_FEAT - above) : 0u;
  }
}

__global__ void hd_compact2(const float4* __restrict__ cand, unsigned* __restrict__ scal,
                            float4* __restrict__ sel, float4* __restrict__ tie)
{
  unsigned n = scal[0]; if (n > CAND_MAX) n = CAND_MAX;
  unsigned T = scal[1];
  for (unsigned i = blockIdx.x * blockDim.x + threadIdx.x; i < n; i += gridDim.x * blockDim.x) {
    float4 c = cand[i];
    unsigned bin = __float_as_uint(c.x) >> 20;
    if (bin > T) {
      unsigned j = atomicAdd(&scal[4], 1u);
      if (j < 2048u) sel[j] = c;
    } else if (bin == T) {
      unsigned j = atomicAdd(&scal[5], 1u);
      if (j < (unsigned)TIE_MAX) tie[j] = c;
    }
  }
}

__device__ inline void hd_bitonic_desc(float* key, int* idx, int n) {
  const int tid = threadIdx.x;
  const int nt  = blockDim.x;
  for (int k = 2; k <= n; k <<= 1) {
    for (int j = k >> 1; j > 0; j >>= 1) {
      for (int i = tid; i < n; i += nt) {
        int ixj = i ^ j;
        if (ixj > i) {
          bool up = ((i & k) == 0);
          float a = key[i], b = key[ixj];
          bool sw = up ? (a < b) : (a > b);
          if (sw) {
            key[i] = b; key[ixj] = a;
            int t = idx[i]; idx[i] = idx[ixj]; idx[ixj] = t;
          }
        }
      }
      __syncthreads();
    }
  }
}

__global__ __launch_bounds__(1024) void hd_final_select(
    const float4* __restrict__ sel, const float4* __restrict__ tie,
    const unsigned* __restrict__ scal, float4* __restrict__ fin, float* __restrict__ out)
{
  __shared__ float sKey[TIE_MAX];
  __shared__ int   sIdx[TIE_MAX];
  const int tid = threadIdx.x;
  unsigned above = scal[2];
  unsigned need  = scal[3];
  unsigned tieN  = scal[5]; if (tieN > (unsigned)TIE_MAX) tieN = TIE_MAX;
  if (above > (unsigned)K_FEAT) above = K_FEAT;

  for (int i = tid; i < TIE_MAX; i += 1024) {
    if (i < (int)tieN) { sKey[i] = tie[i].x; sIdx[i] = i; }
    else               { sKey[i] = -1.f;     sIdx[i] = -1; }
  }
  __syncthreads();
  hd_bitonic_desc(sKey, sIdx, TIE_MAX);

  unsigned take = (need < tieN) ? need : tieN;
  for (int i = tid; i < 2048; i += 1024) {
    float4 e = make_float4(0.f, 0.f, 0.f, 0.f);
    if (i < (int)above)               e = sel[i];
    else if (i < (int)(above + take)) e = tie[sIdx[i - (int)above]];
    fin[i] = e;
  }
  __syncthreads();

  for (int i = tid; i < 2048; i += 1024) { sKey[i] = fin[i].x; sIdx[i] = i; }
  __syncthreads();
  hd_bitonic_desc(sKey, sIdx, 2048);

  for (int k = tid; k < K_FEAT; k += 1024) {
    float4 e = fin[sIdx[k]];
    float s = MR_SIZE * e.w;
    float* L = out + (size_t)k * 6;
    L[0] = s;   L[1] = 0.f; L[2] = e.y;
    L[3] = 0.f; L[4] = s;   L[5] = e.z;
    out[(size_t)K_FEAT * 6 + k] = e.x;
  }
}

// ---------------- host dispatch helpers ----------------
static inline void hd_launch_hblur(const float* s, float* d, int W, int H,
                                   const float* coef, int r, hipStream_t st) {
  dim3 g((W + TX - 1) / TX, (H + 15) / 16);
  switch (r) {
    case 4:  hd_hblur_wmma<4> <<<g, 256, 0, st>>>(s, d, W, H, coef); break;
    case 5:  hd_hblur_wmma<5> <<<g, 256, 0, st>>>(s, d, W, H, coef); break;
    case 6:  hd_hblur_wmma<6> <<<g, 256, 0, st>>>(s, d, W, H, coef); break;
    case 9:  hd_hblur_wmma<9> <<<g, 256, 0, st>>>(s, d, W, H, coef); break;
    default: hd_hblur_wmma<12><<<g, 256, 0, st>>>(s, d, W, H, coef); break;
  }
}
static inline void hd_launch_vblur(const float* s, float* d, int W, int H,
                                   const float* coef, int r, hipStream_t st) {
  dim3 g((W + TX - 1) / TX, (H + 15) / 16);
  switch (r) {
    case 4:  hd_vblur_wmma<4> <<<g, 256, 0, st>>>(s, d, W, H, coef); break;
    case 5:  hd_vblur_wmma<5> <<<g, 256, 0, st>>>(s, d, W, H, coef); break;
    case 6:  hd_vblur_wmma<6> <<<g, 256, 0, st>>>(s, d, W, H, coef); break;
    case 9:  hd_vblur_wmma<9> <<<g, 256, 0, st>>>(s, d, W, H, coef); break;
    default: hd_vblur_wmma<12><<<g, 256, 0, st>>>(s, d, W, H, coef); break;
  }
}

// ---------------- host driver ----------------
extern "C" void kernel_launch(void* const* d_in, const int* in_sizes, int n_in,
                              void* d_out, int out_size, void* d_ws, size_t ws_size,
                              hipStream_t stream) {
  (void)n_in; (void)out_size;
  const float* x = (const float*)d_in[0];
  int W = 2048, H = 2048;
  {
    long long n = in_sizes[0];
    int s = (int)(sqrt((double)n) + 0.5);
    if ((long long)s * s == n) { W = s; H = s; }
  }
  const size_t N0 = (size_t)W * H;

  // workspace layout (floats)
  float* ws    = (float*)d_ws;
  float* tmp   = ws;                    // N0
  float* baseA = ws + N0;               // N0
  float* baseB = ws + 2 * N0;           // N0
  float* lev   = ws + 3 * N0;           // N0
  float* resp  = ws + 4 * N0;           // 5*N0
  float* coef  = ws + 9 * N0;           // 64 (padded)
  float4* cand = (float4*)(ws + 9 * N0 + 64);       // CAND_MAX
  float4* sel  = cand + CAND_MAX;                   // 2048
  float4* tie  = sel + 2048;                        // TIE_MAX
  float4* fin  = tie + TIE_MAX;                     // 2048
  unsigned* scal = (unsigned*)(fin + 2048);         // 16
  unsigned* hist = scal + 16;                       // 2048
  size_t needBytes = (size_t)((unsigned char*)(hist + 2048) - (unsigned char*)d_ws);
  if (ws_size < needBytes) return;

  const float SIG[5] = {1.6f, 2.0158737f, 2.5398417f, 3.2f, 4.0317473f};
  const int   RAD[4] = {4, 6, 9, 12};

  hd_prep_coeffs<<<1, 32, 0, stream>>>(coef);
  hd_init_ws<<<(2048 + 255) / 256, 256, 0, stream>>>(scal, hist);

  // initial blur to base (sigma = 1.6, r = 5)
  hd_launch_hblur(x,   tmp,   W, H, coef + 0 * 32, 5, stream);
  hd_launch_vblur(tmp, baseA, W, H, coef + 0 * 32, 5, stream);

  int n_oct = (int)floor(log2((double)((W < H ? W : H)) / (4.0 * BORDER))) + 1;
  if (n_oct < 1) n_oct = 1;

  float* base = baseA;
  float* nxt  = baseB;
  int w = W, h = H;
  for (int o = 0; o < n_oct; ++o) {
    float pix = (float)(1 << o);
    dim3 g16((w + 15) / 16, (h + 15) / 16);
    {
      double s = SIG[0];
      hd_hess_resp<<<g16, 256, 0, stream>>>(base, resp, w, h, (float)(s * s * s * s));
    }
    for (int i = 1; i <= 4; ++i) {
      hd_launch_hblur(base, tmp, w, h, coef + i * 32, RAD[i - 1], stream);
      hd_launch_vblur(tmp,  lev, w, h, coef + i * 32, RAD[i - 1], stream);
      double s = SIG[i];
      hd_hess_resp<<<g16, 256, 0, stream>>>(lev, resp + (size_t)i * N0, w, h, (float)(s * s * s * s));
      if (i == 3 && (o + 1) < n_oct) {
        dim3 gd((w / 2 + 15) / 16, (h / 2 + 15) / 16);
        hd_downsample<<<gd, 256, 0, stream>>>(lev, nxt, w / 2, h / 2, w);
      }
    }
    int iw = w - 2 * BORDER, ih = h - 2 * BORDER;
    if (iw > 0 && ih > 0) {
      dim3 gi((iw + 15) / 16, (ih + 15) / 16);
      hd_nms_compact<<<gi, 256, 0, stream>>>(resp, N0, w, h, pix,
                                             SIG[1] * pix, SIG[2] * pix, SIG[3] * pix,
                                             cand, &scal[0]);
    }
    float* t2 = base; base = nxt; nxt = t2;
    w >>= 1; h >>= 1;
  }

  hd_hist<<<256, 256, 0, stream>>>(cand, scal, hist);
  hd_thresh<<<1, 32, 0, stream>>>(hist, scal);
  hd_compact2<<<256, 256, 0, stream>>>(cand, scal, sel, tie);
  hd_final_select<<<1, 1024, 0, stream>>>(sel, tie, scal, fin, (float*)d_out);
}